// Model_59837484368153
// MI455X (gfx1250) — compile-verified
//
#include <hip/hip_runtime.h>

typedef __attribute__((ext_vector_type(2))) float v2f;
typedef __attribute__((ext_vector_type(8))) float v8f;

#define NWG 1024
#define TPB 256

__device__ __forceinline__ float lane_bcast(float v, int lane) {
  return __int_as_float(__builtin_amdgcn_readlane(__float_as_int(v), lane));
}

// Full 32-lane f32 sum via V_WMMA_F32_16X16X4_F32.
// A: a0 = x, a1 = 0  -> row m of A holds {x(lane m), 0, x(lane m+16), 0} (some K order).
// B: all ones (4x16) -> D[m][n] = x(m) + x(m+16) for every n (K-order invariant).
// C/D layout: lane L, reg v holds D[v + 8*(L>=16)][L%16]. Sum of the 8 regs on
// lane 0 covers rows 0-7, on lane 16 rows 8-15; their sum is the 32-lane total.
__device__ __forceinline__ float wave_sum_wmma(float x) {
  v2f a; a[0] = x;    a[1] = 0.0f;
  v2f b; b[0] = 1.0f; b[1] = 1.0f;
  v8f c = {};
  v8f d = __builtin_amdgcn_wmma_f32_16x16x4_f32(false, a, false, b, (short)0, c,
                                                false, false);
  float s = ((d[0] + d[1]) + (d[2] + d[3])) + ((d[4] + d[5]) + (d[6] + d[7]));
  return lane_bcast(s, 0) + lane_bcast(s, 16);
}

__device__ __forceinline__ void accum_point(float px, float py,
                                            float tx, float ty,
                                            float v0x, float v0y, float v1x, float v1y,
                                            float v2x, float v2y,
                                            float e0x, float e0y, float e1x, float e1y,
                                            float e2x, float e2y,
                                            float& g, float& cnt) {
  // gaussian(||cam - p||), mu=2 sigma=4; normalization applied once at the end
  float dx = tx - px, dy = ty - py;
  float dist = sqrtf(dx * dx + dy * dy);
  float t = (dist - 2.0f) * 0.25f;
  g += __expf(-0.5f * t * t);
  // strict inside-triangle test (all crosses same sign)
  float c0 = e0x * (py - v0y) - e0y * (px - v0x);
  float c1 = e1x * (py - v1y) - e1y * (px - v1x);
  float c2 = e2x * (py - v2y) - e2y * (px - v2x);
  bool ins = (c0 > 0.0f && c1 > 0.0f && c2 > 0.0f) ||
             (c0 < 0.0f && c1 < 0.0f && c2 < 0.0f);
  cnt += ins ? 1.0f : 0.0f;
}

__global__ __launch_bounds__(TPB) void fov_partial_kernel(
    const float* __restrict__ pts, const float* __restrict__ pose,
    float* __restrict__ ws, int nfloat) {
  const float tx = pose[0], ty = pose[1], yaw = pose[2];
  const float cy = cosf(yaw), sy = sinf(yaw);
  // POLY = [[0,0],[2,7],[-2,7]] rotated by yaw, translated by (tx,ty)
  const float v0x = tx,                      v0y = ty;
  const float v1x =  2.0f * cy - 7.0f * sy + tx, v1y =  2.0f * sy + 7.0f * cy + ty;
  const float v2x = -2.0f * cy - 7.0f * sy + tx, v2y = -2.0f * sy + 7.0f * cy + ty;
  const float e0x = v1x - v0x, e0y = v1y - v0y;
  const float e1x = v2x - v1x, e1y = v2y - v1y;
  const float e2x = v0x - v2x, e2y = v0y - v2y;

  float g = 0.0f, cnt = 0.0f;
  const int n4 = nfloat >> 2;                       // float4 = two (x,y) points
  const float4* __restrict__ p4 = (const float4*)pts;
  const int stride = gridDim.x * blockDim.x;
  for (int i = blockIdx.x * blockDim.x + threadIdx.x; i < n4; i += stride) {
    float4 q = p4[i];                               // global_load_b128
    accum_point(q.x, q.y, tx, ty, v0x, v0y, v1x, v1y, v2x, v2y,
                e0x, e0y, e1x, e1y, e2x, e2y, g, cnt);
    accum_point(q.z, q.w, tx, ty, v0x, v0y, v1x, v1y, v2x, v2y,
                e0x, e0y, e1x, e1y, e2x, e2y, g, cnt);
  }
  // tail (nfloat % 4 == 2 -> one leftover point), done by one thread; reconverges
  if (blockIdx.x == 0 && threadIdx.x == 0) {
    for (int j = n4 << 2; j + 1 < nfloat; j += 2) {
      accum_point(pts[j], pts[j + 1], tx, ty, v0x, v0y, v1x, v1y, v2x, v2y,
                  e0x, e0y, e1x, e1y, e2x, e2y, g, cnt);
    }
  }

  // intra-wave reduction on the matrix pipe (f32 exact), EXEC all-ones here
  float wgsum = wave_sum_wmma(g);
  float wcsum = wave_sum_wmma(cnt);

  __shared__ float sg[TPB / 32];
  __shared__ float sc[TPB / 32];
  const int wave = threadIdx.x >> 5;
  if ((threadIdx.x & 31) == 0) { sg[wave] = wgsum; sc[wave] = wcsum; }
  __syncthreads();
  if (threadIdx.x == 0) {
    float G = 0.0f, C = 0.0f;
    for (int w = 0; w < TPB / 32; ++w) { G += sg[w]; C += sc[w]; }  // fixed order
    ws[2 * blockIdx.x]     = G;
    ws[2 * blockIdx.x + 1] = C;
  }
}

__global__ __launch_bounds__(32) void fov_final_kernel(
    const float* __restrict__ ws, float* __restrict__ out, int nwg) {
  float g = 0.0f, c = 0.0f;
  for (int i = threadIdx.x; i < nwg; i += 32) {   // fixed per-lane order
    g += ws[2 * i];
    c += ws[2 * i + 1];
  }
  float G = wave_sum_wmma(g);
  float C = wave_sum_wmma(c);
  if (threadIdx.x == 0) {
    const float INV_NORM = 0.09973557010f;        // 1/(4*sqrt(2*pi))
    const float EPS = 1e-6f;
    out[0] = 1.0f / (G * INV_NORM + EPS) + 1.0f / (C + EPS);
  }
}

extern "C" void kernel_launch(void* const* d_in, const int* in_sizes, int n_in,
                              void* d_out, int out_size, void* d_ws, size_t ws_size,
                              hipStream_t stream) {
  const float* pts  = (const float*)d_in[0];   // [N,2] float32
  const float* pose = (const float*)d_in[1];   // [3]   float32
  float* ws = (float*)d_ws;

  const int nfloat = in_sizes[0];
  int nwg = NWG;
  const size_t need = (size_t)nwg * 2u * sizeof(float);
  if (need > ws_size) nwg = (int)(ws_size / (2u * sizeof(float)));
  if (nwg < 1) nwg = 1;

  fov_partial_kernel<<<nwg, TPB, 0, stream>>>(pts, pose, ws, nfloat);
  fov_final_kernel<<<1, 32, 0, stream>>>(ws, (float*)d_out, nwg);
}